// GraphFeatureExtractor_89369679495223
// MI455X (gfx1250) — compile-verified
//
#include <hip/hip_runtime.h>
#include <math.h>

typedef __attribute__((ext_vector_type(2))) float v2f;
typedef __attribute__((ext_vector_type(8))) float v8f;

#define NEG_SLOPE 0.2f
#define KMAX 256              // largest K used by any GEMM here
#define LDSK (KMAX + 2)       // pad to avoid 16-way LDS bank conflicts

// ---------------------------------------------------------------- utilities

__global__ __launch_bounds__(256) void fill_kernel(float* __restrict__ p, float v, int n) {
  int i = blockIdx.x * blockDim.x + threadIdx.x;
  if (i < n) p[i] = v;
}

__device__ __forceinline__ float atomicMaxF(float* addr, float val) {
  int* ia = (int*)addr;
  int cur = __float_as_int(*addr);
  while (__int_as_float(cur) < val) {
    int prev = atomicCAS(ia, cur, __float_as_int(val));
    if (prev == cur) break;
    cur = prev;
  }
  return __int_as_float(cur);
}

// ------------------------------------------------ fp32 WMMA GEMM (CDNA5)
// C[M,Nc] = A[M,K] @ W[K,Nc]   (fp32, V_WMMA_F32_16X16X4_F32)
// Requires M%16==0, K%4==0, K<=KMAX, Nc%16==0.
// Block = 256 threads = 8 waves = 8 row-tiles x 1 col-tile (128x16 output).
// The 16-column B panel (K x 16) is staged once per block into LDS,
// transposed so each lane's K-pair is a contiguous ds_load_b64.
// A frag (16x4): lane L holds row M=L%16, K pair = (L<16)?{0,1}:{2,3}
// B frag (4x16): lane L holds col N=L%16, K pair = (L<16)?{0,1}:{2,3}
// C/D frag:      lane L holds col N=L%16, VGPR v holds M = v + 8*(L>=16)
__global__ __launch_bounds__(256) void gemm_wmma_f32(
    const float* __restrict__ A, const float* __restrict__ W,
    float* __restrict__ C, int M, int K, int Nc) {
  __shared__ float ldsB[16 * LDSK];            // 16.5 KB of the 320 KB WGP LDS

  const int tilesM    = M >> 4;
  const int rowBlocks = (tilesM + 7) >> 3;
  const int colTile   = blockIdx.x / rowBlocks;
  const int rowBlk    = blockIdx.x % rowBlocks;

  // Cooperative, coalesced load of the B panel W[0:K, colTile*16 : +16],
  // stored transposed: ldsB[col * LDSK + k].
  for (int idx = threadIdx.x; idx < (K << 4); idx += 256) {
    int k = idx >> 4;
    int c = idx & 15;
    ldsB[c * LDSK + k] = W[(size_t)k * Nc + colTile * 16 + c];
  }
  __syncthreads();

  const int lane = threadIdx.x & 31;
  const int wave = threadIdx.x >> 5;
  const int rowTile = rowBlk * 8 + wave;       // uniform per wave
  if (rowTile >= tilesM) return;               // whole-wave exit after barrier

  const int rc   = lane & 15;                  // A row / B col within tile
  const int kOff = (lane >> 4) << 1;           // 0 or 2

  const float*  aRow  = A + (size_t)(rowTile * 16 + rc) * K;
  const float2* bPair = (const float2*)(ldsB + rc * LDSK);  // 8B-aligned (LDSK even)

  v8f acc = {};
#pragma unroll 4
  for (int k = 0; k < K; k += 4) {
    __builtin_prefetch(aRow + k + 32, 0, 0);   // global_prefetch_b8
    v2f a, b;
    a.x = aRow[k + kOff];                      // contiguous -> global_load_b64
    a.y = aRow[k + kOff + 1];
    float2 bp = bPair[(k + kOff) >> 1];        // contiguous -> ds_load_b64
    b.x = bp.x;
    b.y = bp.y;
    // (neg_a, A, neg_b, B, c_mod, C, reuse_a, reuse_b)
    acc = __builtin_amdgcn_wmma_f32_16x16x4_f32(
        false, a, false, b, (short)0, acc, false, false);
  }

  const int mBase = rowTile * 16 + ((lane >> 4) << 3);
  float* cPtr = C + (size_t)mBase * Nc + colTile * 16 + rc;
#pragma unroll
  for (int v = 0; v < 8; ++v) cPtr[(size_t)v * Nc] = acc[v];
}

// -------------------------------------------------------- edge-phase kernels
// Virtual self-loops: edge id e in [E, E+N) means src=dst=e-E.
// One thread per (edge, head); H == 2 hardcoded.

__global__ __launch_bounds__(256) void edge_logit_max_kernel(
    const float* __restrict__ xl, const float* __restrict__ xr,
    const float* __restrict__ att, const int* __restrict__ srcI,
    const int* __restrict__ dstI, float* __restrict__ logits,
    float* __restrict__ mmax, int E, int N, int C) {
  int t = blockIdx.x * blockDim.x + threadIdx.x;
  int total = (E + N) * 2;
  if (t >= total) return;
  int e = t >> 1, h = t & 1;
  int s, d;
  if (e < E) { s = srcI[e]; d = dstI[e]; } else { s = d = e - E; }
  const int HC = C * 2;
  const float4* xlv = (const float4*)(xl + (size_t)s * HC + h * C);
  const float4* xrv = (const float4*)(xr + (size_t)d * HC + h * C);
  const float4* av  = (const float4*)(att + h * C);
  float dot = 0.f;
  for (int c = 0; c < (C >> 2); ++c) {
    float4 a4 = xlv[c], b4 = xrv[c], w4 = av[c];
    float v0 = a4.x + b4.x; v0 = v0 > 0.f ? v0 : NEG_SLOPE * v0;
    float v1 = a4.y + b4.y; v1 = v1 > 0.f ? v1 : NEG_SLOPE * v1;
    float v2 = a4.z + b4.z; v2 = v2 > 0.f ? v2 : NEG_SLOPE * v2;
    float v3 = a4.w + b4.w; v3 = v3 > 0.f ? v3 : NEG_SLOPE * v3;
    dot += v0 * w4.x + v1 * w4.y + v2 * w4.z + v3 * w4.w;
  }
  logits[t] = dot;
  atomicMaxF(&mmax[d * 2 + h], dot);
}

__global__ __launch_bounds__(256) void edge_exp_sum_kernel(
    float* __restrict__ logits, const float* __restrict__ mmax,
    float* __restrict__ denom, const int* __restrict__ dstI, int E, int N) {
  int t = blockIdx.x * blockDim.x + threadIdx.x;
  int total = (E + N) * 2;
  if (t >= total) return;
  int e = t >> 1, h = t & 1;
  int d = (e < E) ? dstI[e] : e - E;
  float a = __expf(logits[t] - mmax[d * 2 + h]);
  logits[t] = a;
  atomicAdd(&denom[d * 2 + h], a);
}

__global__ __launch_bounds__(256) void edge_aggregate_kernel(
    const float* __restrict__ xl, const float* __restrict__ logits,
    const float* __restrict__ denom, const int* __restrict__ srcI,
    const int* __restrict__ dstI, float* __restrict__ out,
    int E, int N, int C) {
  int t = blockIdx.x * blockDim.x + threadIdx.x;
  int total = (E + N) * 2;
  if (t >= total) return;
  int e = t >> 1, h = t & 1;
  int s, d;
  if (e < E) { s = srcI[e]; d = dstI[e]; } else { s = d = e - E; }
  float alpha = logits[t] / denom[d * 2 + h];
  const int HC = C * 2;
  const float4* xv = (const float4*)(xl + (size_t)s * HC + h * C);
  float* orow = out + (size_t)d * HC + h * C;
  for (int c = 0; c < (C >> 2); ++c) {
    float4 v = xv[c];
    atomicAdd(&orow[c * 4 + 0], alpha * v.x);
    atomicAdd(&orow[c * 4 + 1], alpha * v.y);
    atomicAdd(&orow[c * 4 + 2], alpha * v.z);
    atomicAdd(&orow[c * 4 + 3], alpha * v.w);
  }
}

__global__ __launch_bounds__(256) void relu_bias_kernel(
    float* __restrict__ p, const float* __restrict__ bias, int n, int hc) {
  int i = blockIdx.x * blockDim.x + threadIdx.x;
  if (i < n) {
    float v = p[i] + bias[i % hc];
    p[i] = v > 0.f ? v : 0.f;
  }
}

__global__ __launch_bounds__(256) void bias_add_kernel(
    float* __restrict__ p, const float* __restrict__ bias, int n, int hc) {
  int i = blockIdx.x * blockDim.x + threadIdx.x;
  if (i < n) p[i] += bias[i % hc];
}

// ---------------------------------------------------------------- launcher

extern "C" void kernel_launch(void* const* d_in, const int* in_sizes, int n_in,
                              void* d_out, int out_size, void* d_ws, size_t ws_size,
                              hipStream_t stream) {
  (void)n_in; (void)out_size; (void)ws_size;
  const float* x    = (const float*)d_in[0];
  const int*   ei   = (const int*)d_in[1];
  const float* Wl0  = (const float*)d_in[2];
  const float* Wr0  = (const float*)d_in[3];
  const float* att0 = (const float*)d_in[4];
  const float* b0   = (const float*)d_in[5];
  const float* Wl1  = (const float*)d_in[6];
  const float* Wr1  = (const float*)d_in[7];
  const float* att1 = (const float*)d_in[8];
  const float* b1   = (const float*)d_in[9];
  float* out = (float*)d_out;

  const int CIN = 256, C1 = 64, C2 = 128;
  const int HC0 = 2 * C1;   // 128
  const int HC1 = 2 * C2;   // 256
  const int N = in_sizes[0] / CIN;
  const int E = in_sizes[1] / 2;
  const int* srcI = ei;
  const int* dstI = ei + E;

  // workspace layout (floats)
  float* ws = (float*)d_ws;
  size_t off = 0;
  float* xl0   = ws + off; off += (size_t)N * HC0;
  float* xr0   = ws + off; off += (size_t)N * HC0;
  float* hbuf  = ws + off; off += (size_t)N * HC0;   // agg0, then h in-place
  float* xl1   = ws + off; off += (size_t)N * HC1;
  float* xr1   = ws + off; off += (size_t)N * HC1;
  float* logit = ws + off; off += (size_t)(E + N) * 2;
  float* mmax  = ws + off; off += (size_t)N * 2;
  float* denom = ws + off; off += (size_t)N * 2;

  const int T = 256;
  const int ET2 = (E + N) * 2;
  const int eBlk = (ET2 + T - 1) / T;
  const int nh2  = N * 2;
  const int nhB  = (nh2 + T - 1) / T;

  // ---------------- layer 0 linear transforms (WMMA fp32) ----------------
  {
    int rowBlocks = ((N >> 4) + 7) >> 3;
    int blk = rowBlocks * (HC0 >> 4);
    gemm_wmma_f32<<<blk, T, 0, stream>>>(x, Wl0, xl0, N, CIN, HC0);
    gemm_wmma_f32<<<blk, T, 0, stream>>>(x, Wr0, xr0, N, CIN, HC0);
  }

  // ---------------- layer 0 attention + segment softmax ------------------
  fill_kernel<<<nhB, T, 0, stream>>>(mmax,  -INFINITY, nh2);
  fill_kernel<<<nhB, T, 0, stream>>>(denom, 0.f,       nh2);
  edge_logit_max_kernel<<<eBlk, T, 0, stream>>>(xl0, xr0, att0, srcI, dstI,
                                                logit, mmax, E, N, C1);
  edge_exp_sum_kernel<<<eBlk, T, 0, stream>>>(logit, mmax, denom, dstI, E, N);
  {
    int n0 = N * HC0;
    fill_kernel<<<(n0 + T - 1) / T, T, 0, stream>>>(hbuf, 0.f, n0);
  }
  edge_aggregate_kernel<<<eBlk, T, 0, stream>>>(xl0, logit, denom, srcI, dstI,
                                                hbuf, E, N, C1);
  {
    int n0 = N * HC0;
    relu_bias_kernel<<<(n0 + T - 1) / T, T, 0, stream>>>(hbuf, b0, n0, HC0);
  }

  // ---------------- layer 1 linear transforms (WMMA fp32) ----------------
  {
    int rowBlocks = ((N >> 4) + 7) >> 3;
    int blk = rowBlocks * (HC1 >> 4);
    gemm_wmma_f32<<<blk, T, 0, stream>>>(hbuf, Wl1, xl1, N, HC0, HC1);
    gemm_wmma_f32<<<blk, T, 0, stream>>>(hbuf, Wr1, xr1, N, HC0, HC1);
  }

  // ---------------- layer 1 attention + segment softmax ------------------
  fill_kernel<<<nhB, T, 0, stream>>>(mmax,  -INFINITY, nh2);
  fill_kernel<<<nhB, T, 0, stream>>>(denom, 0.f,       nh2);
  edge_logit_max_kernel<<<eBlk, T, 0, stream>>>(xl1, xr1, att1, srcI, dstI,
                                                logit, mmax, E, N, C2);
  edge_exp_sum_kernel<<<eBlk, T, 0, stream>>>(logit, mmax, denom, dstI, E, N);
  {
    int n1 = N * HC1;
    fill_kernel<<<(n1 + T - 1) / T, T, 0, stream>>>(out, 0.f, n1);
  }
  edge_aggregate_kernel<<<eBlk, T, 0, stream>>>(xl1, logit, denom, srcI, dstI,
                                                out, E, N, C2);
  {
    int n1 = N * HC1;
    bias_add_kernel<<<(n1 + T - 1) / T, T, 0, stream>>>(out, b1, n1, HC1);
  }
}